// ODEFunc_10986526343306
// MI455X (gfx1250) — compile-verified
//
#include <hip/hip_runtime.h>

typedef float v2f __attribute__((ext_vector_type(2)));
typedef float v8f __attribute__((ext_vector_type(8)));

#define HID 64
#define LN_EPS 1e-5f
#define ROWS_PER_BLOCK 128
#define SA_STRIDE 65   // 64 + 1 pad: row stride odd -> consecutive rows land on distinct LDS banks

// ---------------------------------------------------------------- degree pass
__global__ void set_ones_kernel(float* __restrict__ p, int n) {
    int i = blockIdx.x * blockDim.x + threadIdx.x;
    if (i < n) p[i] = 1.0f;  // self-loop contributes 1 to every node's degree
}

__global__ void count_deg_kernel(const long long* __restrict__ edges, long long E,
                                 float* __restrict__ deg) {
    long long i = (long long)blockIdx.x * blockDim.x + threadIdx.x;
    if (i < E) {
        int dst = (int)edges[E + i];          // edge_index[1]
        atomicAdd(&deg[dst], 1.0f);           // exact: counts < 2^24
    }
}

__global__ void finalize_dinv_kernel(float* __restrict__ d, int n) {
    int i = blockIdx.x * blockDim.x + threadIdx.x;
    if (i < n) d[i] = rsqrtf(d[i]);           // deg >= 1 always (self loops)
}

// ------------------------------------------------- layernorm + x = hn @ Wconv
__global__ __launch_bounds__(256) void ln_gemm_kernel(
    const float* __restrict__ h, const float* __restrict__ gamma,
    const float* __restrict__ beta, const float* __restrict__ W,
    float* __restrict__ xout, int N)
{
    __shared__ float sW[HID * HID];                      // 16 KB, row-major W[k][j]
    __shared__ float sA[ROWS_PER_BLOCK * SA_STRIDE];     // ~33 KB, hn tile

    const int t = threadIdx.x;
    const int rowBase = blockIdx.x * ROWS_PER_BLOCK;

    // load W into LDS (4096 floats / 256 threads)
#pragma unroll
    for (int i = 0; i < (HID * HID) / 256; ++i)
        sW[t + i * 256] = W[t + i * 256];

    // LayerNorm: 2 threads per row, 32 contiguous elems each, pair-reduce via shfl_xor
    {
        const int r    = t >> 1;
        const int half = t & 1;
        const int grow = rowBase + r;
        float vals[32];
        float s = 0.f, ss = 0.f;
        if (grow < N) {
            const float4* hp = (const float4*)(h + (size_t)grow * HID + half * 32);
#pragma unroll
            for (int j = 0; j < 8; ++j) {
                float4 v = hp[j];
                vals[4*j+0] = v.x; vals[4*j+1] = v.y; vals[4*j+2] = v.z; vals[4*j+3] = v.w;
                s  += v.x + v.y + v.z + v.w;
                ss += v.x*v.x + v.y*v.y + v.z*v.z + v.w*v.w;
            }
        } else {
#pragma unroll
            for (int j = 0; j < 32; ++j) vals[j] = 0.f;
        }
        s  += __shfl_xor(s, 1);
        ss += __shfl_xor(ss, 1);
        const float mu  = s * (1.0f / HID);
        const float var = ss * (1.0f / HID) - mu * mu;
        const float inv = rsqrtf(var + LN_EPS);
#pragma unroll
        for (int j = 0; j < 32; ++j) {
            const int f = half * 32 + j;
            const float o = (grow < N) ? ((vals[j] - mu) * inv * gamma[f] + beta[f]) : 0.f;
            sA[r * SA_STRIDE + f] = o;
        }
    }
    __syncthreads();

    // WMMA: wave w handles a 16x64 output stripe; K=64 in 16 steps of 4 (fp32 WMMA)
    const int wave  = t >> 5;
    const int lane  = t & 31;
    const int rbase = wave * 16;
    const int lrow  = lane & 15;
    const int su    = lane >> 4;          // half-wave select (K/row split per ISA layout)
    v8f acc0 = {}, acc1 = {}, acc2 = {}, acc3 = {};
#pragma unroll
    for (int kb = 0; kb < 16; ++kb) {
        const int k0 = kb * 4 + su * 2;
        v2f a;
        a.x = sA[(rbase + lrow) * SA_STRIDE + k0];
        a.y = sA[(rbase + lrow) * SA_STRIDE + k0 + 1];
        v2f b0, b1, b2, b3;
        b0.x = sW[(k0    ) * HID +  0 + lrow]; b0.y = sW[(k0 + 1) * HID +  0 + lrow];
        b1.x = sW[(k0    ) * HID + 16 + lrow]; b1.y = sW[(k0 + 1) * HID + 16 + lrow];
        b2.x = sW[(k0    ) * HID + 32 + lrow]; b2.y = sW[(k0 + 1) * HID + 32 + lrow];
        b3.x = sW[(k0    ) * HID + 48 + lrow]; b3.y = sW[(k0 + 1) * HID + 48 + lrow];
        acc0 = __builtin_amdgcn_wmma_f32_16x16x4_f32(false, a, false, b0, (short)0, acc0, false, false);
        acc1 = __builtin_amdgcn_wmma_f32_16x16x4_f32(false, a, false, b1, (short)0, acc1, false, false);
        acc2 = __builtin_amdgcn_wmma_f32_16x16x4_f32(false, a, false, b2, (short)0, acc2, false, false);
        acc3 = __builtin_amdgcn_wmma_f32_16x16x4_f32(false, a, false, b3, (short)0, acc3, false, false);
    }
    // D layout: VGPR r -> rows su*8 + r, cols = lane%16 within each 16-col tile
    const int gm = rowBase + rbase + su * 8;
#pragma unroll
    for (int r = 0; r < 8; ++r) {
        const int m = gm + r;
        if (m < N) {
            float* o = xout + (size_t)m * HID;
            o[ 0 + lrow] = acc0[r];
            o[16 + lrow] = acc1[r];
            o[32 + lrow] = acc2[r];
            o[48 + lrow] = acc3[r];
        }
    }
}

// ------------------------------------------------------- self-loop term + scatter
__global__ void init_agg_kernel(const float* __restrict__ x, const float* __restrict__ dinv,
                                float* __restrict__ agg, int total) {
    int i = blockIdx.x * blockDim.x + threadIdx.x;
    if (i < total) {
        const float di = dinv[i >> 6];
        agg[i] = x[i] * di * di;      // coef for self loop = dinv[i]^2
    }
}

__global__ void scatter_kernel(const long long* __restrict__ edges, long long E,
                               const float* __restrict__ dinv,
                               const float* __restrict__ x,
                               float* __restrict__ agg)
{
    long long idx = (long long)blockIdx.x * blockDim.x + threadIdx.x;
    long long e = idx >> 4;           // 16 threads per edge, 4 features each
    int q = (int)(idx & 15);
    if (e < E) {
        const int src = (int)edges[e];
        const int dst = (int)edges[E + e];
        const float coef = dinv[src] * dinv[dst];
        const float4 v = *(const float4*)(x + (size_t)src * HID + q * 4);
        float* o = agg + (size_t)dst * HID + q * 4;
        atomicAdd(o + 0, coef * v.x);
        atomicAdd(o + 1, coef * v.y);
        atomicAdd(o + 2, coef * v.z);
        atomicAdd(o + 3, coef * v.w);
    }
}

// ------------------------------- out (+)= (agg + b_conv) @ Wout + bout, clip on pass 1
__global__ __launch_bounds__(256) void out_gemm_kernel(
    const float* __restrict__ agg, const float* __restrict__ bconv,
    const float* __restrict__ W, const float* __restrict__ bout,
    float* __restrict__ out, int N, int pass)
{
    __shared__ float sW[HID * HID];
    __shared__ float sA[ROWS_PER_BLOCK * SA_STRIDE];

    const int t = threadIdx.x;
    const int rowBase = blockIdx.x * ROWS_PER_BLOCK;

#pragma unroll
    for (int i = 0; i < (HID * HID) / 256; ++i)
        sW[t + i * 256] = W[t + i * 256];

    // tile fill: A = agg + b_conv (row-constant bias folded into A)
    {
        const int r    = t >> 1;
        const int half = t & 1;
        const int grow = rowBase + r;
        if (grow < N) {
            const float4* ap = (const float4*)(agg + (size_t)grow * HID + half * 32);
            const float4* bp = (const float4*)(bconv + half * 32);
#pragma unroll
            for (int j = 0; j < 8; ++j) {
                float4 v = ap[j], b = bp[j];
                float* d = &sA[r * SA_STRIDE + half * 32 + 4 * j];
                d[0] = v.x + b.x; d[1] = v.y + b.y; d[2] = v.z + b.z; d[3] = v.w + b.w;
            }
        } else {
#pragma unroll
            for (int j = 0; j < 32; ++j)
                sA[r * SA_STRIDE + half * 32 + j] = 0.f;
        }
    }
    __syncthreads();

    const int wave  = t >> 5;
    const int lane  = t & 31;
    const int rbase = wave * 16;
    const int lrow  = lane & 15;
    const int su    = lane >> 4;
    v8f acc0 = {}, acc1 = {}, acc2 = {}, acc3 = {};
#pragma unroll
    for (int kb = 0; kb < 16; ++kb) {
        const int k0 = kb * 4 + su * 2;
        v2f a;
        a.x = sA[(rbase + lrow) * SA_STRIDE + k0];
        a.y = sA[(rbase + lrow) * SA_STRIDE + k0 + 1];
        v2f b0, b1, b2, b3;
        b0.x = sW[(k0    ) * HID +  0 + lrow]; b0.y = sW[(k0 + 1) * HID +  0 + lrow];
        b1.x = sW[(k0    ) * HID + 16 + lrow]; b1.y = sW[(k0 + 1) * HID + 16 + lrow];
        b2.x = sW[(k0    ) * HID + 32 + lrow]; b2.y = sW[(k0 + 1) * HID + 32 + lrow];
        b3.x = sW[(k0    ) * HID + 48 + lrow]; b3.y = sW[(k0 + 1) * HID + 48 + lrow];
        acc0 = __builtin_amdgcn_wmma_f32_16x16x4_f32(false, a, false, b0, (short)0, acc0, false, false);
        acc1 = __builtin_amdgcn_wmma_f32_16x16x4_f32(false, a, false, b1, (short)0, acc1, false, false);
        acc2 = __builtin_amdgcn_wmma_f32_16x16x4_f32(false, a, false, b2, (short)0, acc2, false, false);
        acc3 = __builtin_amdgcn_wmma_f32_16x16x4_f32(false, a, false, b3, (short)0, acc3, false, false);
    }

    const float bo0 = bout[ 0 + lrow], bo1 = bout[16 + lrow];
    const float bo2 = bout[32 + lrow], bo3 = bout[48 + lrow];
    const int gm = rowBase + rbase + su * 8;
#pragma unroll
    for (int r = 0; r < 8; ++r) {
        const int m = gm + r;
        if (m < N) {
            float* o = out + (size_t)m * HID;
            float y0 = acc0[r] + bo0, y1 = acc1[r] + bo1;
            float y2 = acc2[r] + bo2, y3 = acc3[r] + bo3;
            if (pass == 0) {
                o[ 0 + lrow] = y0; o[16 + lrow] = y1;
                o[32 + lrow] = y2; o[48 + lrow] = y3;
            } else {
                y0 += o[ 0 + lrow]; y1 += o[16 + lrow];
                y2 += o[32 + lrow]; y3 += o[48 + lrow];
                o[ 0 + lrow] = fminf(50.f, fmaxf(-50.f, y0));
                o[16 + lrow] = fminf(50.f, fmaxf(-50.f, y1));
                o[32 + lrow] = fminf(50.f, fmaxf(-50.f, y2));
                o[48 + lrow] = fminf(50.f, fmaxf(-50.f, y3));
            }
        }
    }
}

// ---------------------------------------------------------------- launcher
extern "C" void kernel_launch(void* const* d_in, const int* in_sizes, int n_in,
                              void* d_out, int out_size, void* d_ws, size_t ws_size,
                              hipStream_t stream) {
    const float* h     = (const float*)d_in[1];
    const long long* edges[2] = { (const long long*)d_in[2], (const long long*)d_in[3] };
    const float* gamma = (const float*)d_in[4];
    const float* beta  = (const float*)d_in[5];
    const float* Wconv[2] = { (const float*)d_in[6],  (const float*)d_in[8]  };
    const float* bconv[2] = { (const float*)d_in[7],  (const float*)d_in[9]  };
    const float* Wout[2]  = { (const float*)d_in[10], (const float*)d_in[12] };
    const float* bout[2]  = { (const float*)d_in[11], (const float*)d_in[13] };

    const int N = in_sizes[1] / HID;
    const long long E[2] = { (long long)in_sizes[2] / 2, (long long)in_sizes[3] / 2 };

    // workspace layout (floats): x[N*64] | agg[N*64] | dinv[N]
    float* x    = (float*)d_ws;
    float* agg  = x + (size_t)N * HID;
    float* dinv = agg + (size_t)N * HID;
    float* outp = (float*)d_out;

    const int rowBlocks = (N + ROWS_PER_BLOCK - 1) / ROWS_PER_BLOCK;
    const int nodeBlocks = (N + 255) / 256;
    const int elemBlocks = (N * HID + 255) / 256;

    for (int p = 0; p < 2; ++p) {
        const long long nE = E[p];
        const int edgeBlocks = (int)((nE + 255) / 256);
        const int scatBlocks = (int)((nE * 16 + 255) / 256);

        set_ones_kernel<<<nodeBlocks, 256, 0, stream>>>(dinv, N);
        count_deg_kernel<<<edgeBlocks, 256, 0, stream>>>(edges[p], nE, dinv);
        finalize_dinv_kernel<<<nodeBlocks, 256, 0, stream>>>(dinv, N);
        ln_gemm_kernel<<<rowBlocks, 256, 0, stream>>>(h, gamma, beta, Wconv[p], x, N);
        init_agg_kernel<<<elemBlocks, 256, 0, stream>>>(x, dinv, agg, N * HID);
        scatter_kernel<<<scatBlocks, 256, 0, stream>>>(edges[p], nE, dinv, x, agg);
        out_gemm_kernel<<<rowBlocks, 256, 0, stream>>>(agg, bconv[p], Wout[p], bout[p], outp, N, p);
    }
}